// Pixel_RNNcontrol_75453985456217
// MI455X (gfx1250) — compile-verified
//
#include <hip/hip_runtime.h>

// ---------------------------------------------------------------------------
// Pixel_RNNcontrol: 128-step tanh RNN scan (H=I=2) + sigmoid(10*h).
// Memory-bound (~514 MB traffic -> ~22us @ 23.3 TB/s). One thread owns 2
// batch lanes (16B/step). Deep async-to-LDS prefetch pipeline (ASYNCcnt) to
// keep ~16MB in flight; v_tanh_f32 for tanh and sigmoid (via tanh identity);
// nontemporal b128 stores for the write-once output stream.
// ---------------------------------------------------------------------------

#define BLOCK 256
#define BPT   2            // batch elements per thread (2 floats each -> 16B)
#define DEPTH 8            // outstanding async tile loads per wave
#define NBUF  (DEPTH + 1)  // ring buffers: write slot == slot read last iter

typedef __attribute__((ext_vector_type(4))) float v4f;
typedef __attribute__((ext_vector_type(4))) int   v4i;

#if __has_builtin(__builtin_amdgcn_global_load_async_to_lds_b128) && \
    __has_builtin(__builtin_amdgcn_s_wait_asynccnt)
#define USE_ASYNC 1
#else
#define USE_ASYNC 0
#endif

#if __has_builtin(__builtin_amdgcn_tanhf)
#define FAST_TANH(v) __builtin_amdgcn_tanhf(v)
#else
#define FAST_TANH(v) tanhf(v)
#endif

#if USE_ASYNC
__device__ __forceinline__ void async_tile16(const float* g, v4f* lds) {
  // GLOBAL_LOAD_ASYNC_TO_LDS_B128: per-lane 16B global -> LDS, ASYNCcnt-tracked
  // builtin signature: (int4 AS1* src_global, int4 AS3* dst_lds, imm off, imm cpol)
  __builtin_amdgcn_global_load_async_to_lds_b128(
      (__attribute__((address_space(1))) v4i*)g,
      (__attribute__((address_space(3))) v4i*)lds,
      /*offset=*/0, /*cpol=*/0);
}
#endif

__global__ __launch_bounds__(BLOCK)
void rnn_scan_kernel(const float* __restrict__ x,    // (T, B, 2)
                     const float* __restrict__ h0,   // (1, B, 2)
                     const float* __restrict__ Wih,  // (2, 2) row-major
                     const float* __restrict__ Whh,  // (2, 2) row-major
                     const float* __restrict__ bih,  // (2,)
                     const float* __restrict__ bhh,  // (2,)
                     float* __restrict__ out,        // (T, B, 2)
                     float* __restrict__ hid,        // (B, 2) == out + T*B*2
                     int B, int T)
{
#if USE_ASYNC
  __shared__ v4f xbuf[NBUF][BLOCK];
#endif
  const int tid = threadIdx.x;
  const long long gb = (long long)(blockIdx.x * BLOCK + tid) * BPT;
  if (gb + BPT > (long long)B) return;

  const long long stride = 2LL * (long long)B;  // floats per timestep plane

  // h_t[j] = tanh( sum_k x[k]*Wih[j][k] + sum_k h[k]*Whh[j][k] + b[j] )
  const float w00 = Wih[0], w01 = Wih[1], w10 = Wih[2], w11 = Wih[3];
  const float u00 = Whh[0], u01 = Whh[1], u10 = Whh[2], u11 = Whh[3];
  const float bb0 = bih[0] + bhh[0];
  const float bb1 = bih[1] + bhh[1];

  // state for 2 batch lanes: (h.x,h.y) = lane gb, (h.z,h.w) = lane gb+1
  v4f h = *(const v4f*)(h0 + 2 * gb);

  const float* xcol = x + 2 * gb;
  float* ocol = out + 2 * gb;

  // one RNN step + fused sigmoid(10h) = 0.5 + 0.5*tanh(5h), nontemporal store
  auto step = [&](const v4f xv, long long tt) {
    float p0 = fmaf(u01, h.y, fmaf(u00, h.x, fmaf(w01, xv.y, fmaf(w00, xv.x, bb0))));
    float p1 = fmaf(u11, h.y, fmaf(u10, h.x, fmaf(w11, xv.y, fmaf(w10, xv.x, bb1))));
    float p2 = fmaf(u01, h.w, fmaf(u00, h.z, fmaf(w01, xv.w, fmaf(w00, xv.z, bb0))));
    float p3 = fmaf(u11, h.w, fmaf(u10, h.z, fmaf(w11, xv.w, fmaf(w10, xv.z, bb1))));
    h.x = FAST_TANH(p0);
    h.y = FAST_TANH(p1);
    h.z = FAST_TANH(p2);
    h.w = FAST_TANH(p3);
    v4f o;
    o.x = fmaf(0.5f, FAST_TANH(5.0f * h.x), 0.5f);
    o.y = fmaf(0.5f, FAST_TANH(5.0f * h.y), 0.5f);
    o.z = fmaf(0.5f, FAST_TANH(5.0f * h.z), 0.5f);
    o.w = fmaf(0.5f, FAST_TANH(5.0f * h.w), 0.5f);
    __builtin_nontemporal_store(o, (v4f*)(ocol + tt * stride));
  };

  int t = 0;
  const int tmain = T - DEPTH;

#if USE_ASYNC
  // prologue: fill the pipeline with DEPTH tile loads
  {
    const int pre = (DEPTH < T) ? DEPTH : T;
    for (int p = 0; p < pre; ++p)
      async_tile16(xcol + (long long)p * stride, &xbuf[p][tid]);
  }
  int rb = 0;             // read slot  = t % NBUF
  int wb = DEPTH % NBUF;  // write slot = (t+DEPTH) % NBUF == (t-1) % NBUF (safe)

  // steady state: DEPTH outstanding; wait until load t done (<= DEPTH-1 newer)
  for (; t < tmain; ++t) {
    __builtin_amdgcn_s_wait_asynccnt(DEPTH - 1);
    v4f xv = xbuf[rb][tid];
    async_tile16(xcol + (long long)(t + DEPTH) * stride, &xbuf[wb][tid]);
    rb = (rb + 1 == NBUF) ? 0 : rb + 1;
    wb = (wb + 1 == NBUF) ? 0 : wb + 1;
    step(xv, t);
  }
  // drain: everything already requested; wait-all is cheap here
  for (; t < T; ++t) {
    __builtin_amdgcn_s_wait_asynccnt(0);
    v4f xv = xbuf[rb][tid];
    rb = (rb + 1 == NBUF) ? 0 : rb + 1;
    step(xv, t);
  }
#else
  // fallback: register double-buffer + hardware prefetch DEPTH ahead
  v4f xnext = {};
  if (T > 0) xnext = __builtin_nontemporal_load((const v4f*)xcol);
  for (; t < T; ++t) {
    v4f xv = xnext;
    if (t + 1 < T)
      xnext = __builtin_nontemporal_load((const v4f*)(xcol + (long long)(t + 1) * stride));
    if (t + DEPTH < T)
      __builtin_prefetch(xcol + (long long)(t + DEPTH) * stride, 0, 3);
    step(xv, t);
  }
#endif

  // final hidden state (1, B, 2), appended after out in d_out
  *(v4f*)(hid + 2 * gb) = h;
}

extern "C" void kernel_launch(void* const* d_in, const int* in_sizes, int n_in,
                              void* d_out, int out_size, void* d_ws, size_t ws_size,
                              hipStream_t stream) {
  const float* x   = (const float*)d_in[0];
  const float* h0  = (const float*)d_in[1];
  const float* Wih = (const float*)d_in[2];
  const float* Whh = (const float*)d_in[3];
  const float* bih = (const float*)d_in[4];
  const float* bhh = (const float*)d_in[5];
  float* out = (float*)d_out;

  const int B = in_sizes[1] / 2;        // h: (1, B, 2)
  const int T = in_sizes[0] / (2 * B);  // x: (T, B, 2)
  float* hid = out + (long long)T * (long long)B * 2LL;  // hidden after out

  const int per_block = BLOCK * BPT;
  const int blocks = (B + per_block - 1) / per_block;
  rnn_scan_kernel<<<blocks, BLOCK, 0, stream>>>(x, h0, Wih, Whh, bih, bhh,
                                                out, hid, B, T);
}